// ICP_66073776881817
// MI455X (gfx1250) — compile-verified
//
#include <hip/hip_runtime.h>
#include <hip/hip_bf16.h>

typedef float v2f __attribute__((ext_vector_type(2)));
typedef float v8f __attribute__((ext_vector_type(8)));

#define HH 1080
#define WW 1920
#define NPIX (HH * WW)

#define ACC_BLOCKS 512
#define ACC_THREADS 256
#define WAVES_PER_BLOCK (ACC_THREADS / 32)
#define NSLOTS (ACC_BLOCKS * WAVES_PER_BLOCK)   // 4096 per-wave partial Gram slots

// workspace layout (float words)
#define POSE_OFF   0            // 16 floats: current 4x4 pose
#define GRAM_OFF   16           // 49 floats: reduced 7x7 Gram
#define MINMAX_OFF 72           // 2 uints: min/max bits of depth1
#define NORM_OFF   128          // NPIX*3 floats: normal1
#define PART_OFF   (NORM_OFF + NPIX * 3)  // NSLOTS*49 floats: per-wave partials

__device__ __forceinline__ int clampi(int x, int lo, int hi) {
    return x < lo ? lo : (x > hi ? hi : x);
}

// ---------------------------------------------------------------- init ----
__global__ void icp_init_kernel(const float* __restrict__ pose_in, float* ws) {
    int i = threadIdx.x;
    if (i < 16) ws[POSE_OFF + i] = pose_in[i];
    if (i == 0) {
        ((unsigned*)ws)[MINMAX_OFF]     = 0x7F800000u; // +inf bits
        ((unsigned*)ws)[MINMAX_OFF + 1] = 0u;
    }
}

// ------------------------------------------------------------- min/max ----
__global__ void icp_minmax_kernel(const float* __restrict__ depth1, unsigned* minmax) {
    __shared__ unsigned smin[256], smax[256];
    unsigned lmin = 0x7F800000u, lmax = 0u;
    int stride = gridDim.x * blockDim.x;
    for (int idx = blockIdx.x * blockDim.x + threadIdx.x; idx < NPIX; idx += stride) {
        unsigned b = __float_as_uint(depth1[idx]);  // depth > 0 -> uint order == float order
        lmin = lmin < b ? lmin : b;
        lmax = lmax > b ? lmax : b;
    }
    smin[threadIdx.x] = lmin;
    smax[threadIdx.x] = lmax;
    __syncthreads();
    for (int s = 128; s > 0; s >>= 1) {
        if ((int)threadIdx.x < s) {
            unsigned a = smin[threadIdx.x + s];
            if (a < smin[threadIdx.x]) smin[threadIdx.x] = a;
            unsigned b = smax[threadIdx.x + s];
            if (b > smax[threadIdx.x]) smax[threadIdx.x] = b;
        }
        __syncthreads();
    }
    if (threadIdx.x == 0) {
        atomicMin(&minmax[0], smin[0]);
        atomicMax(&minmax[1], smax[0]);
    }
}

// ------------------------------------------------------------- normals ----
__global__ void icp_normal_kernel(const float* __restrict__ depth1,
                                  const float* __restrict__ Kmat,
                                  const unsigned* __restrict__ minmax,
                                  float* __restrict__ normal1) {
    int pix = blockIdx.x * blockDim.x + threadIdx.x;
    if (pix >= NPIX) return;
    int j = pix / WW, i = pix - j * WW;
    float fx = Kmat[0], cx = Kmat[2], fy = Kmat[4], cy = Kmat[5];

    // Sobel on vertex map with edge-clamped 3x3 neighborhood
    const float wdx[3][3] = {{-1.f, 0.f, 1.f}, {-2.f, 0.f, 2.f}, {-1.f, 0.f, 1.f}};
    const float wdy[3][3] = {{-1.f, -2.f, -1.f}, {0.f, 0.f, 0.f}, {1.f, 2.f, 1.f}};
    float dxv0 = 0.f, dxv1 = 0.f, dxv2 = 0.f;
    float dyv0 = 0.f, dyv1 = 0.f, dyv2 = 0.f;
#pragma unroll
    for (int a = -1; a <= 1; ++a) {
#pragma unroll
        for (int b = -1; b <= 1; ++b) {
            int jj = clampi(j + a, 0, HH - 1);
            int ii = clampi(i + b, 0, WW - 1);
            float d  = depth1[jj * WW + ii];
            float px = ((float)ii - cx) / fx * d;
            float py = ((float)jj - cy) / fy * d;
            float pz = d;
            float wa = wdx[a + 1][b + 1], wb = wdy[a + 1][b + 1];
            dxv0 += wa * px; dxv1 += wa * py; dxv2 += wa * pz;
            dyv0 += wb * px; dyv1 += wb * py; dyv2 += wb * pz;
        }
    }
    // n = cross(dx, dy), normalized
    float nx = dxv1 * dyv2 - dxv2 * dyv1;
    float ny = dxv2 * dyv0 - dxv0 * dyv2;
    float nz = dxv0 * dyv1 - dxv1 * dyv0;
    float inv = 1.0f / (sqrtf(nx * nx + ny * ny + nz * nz) + 1e-8f);
    nx *= inv; ny *= inv; nz *= inv;

    float z = depth1[pix];
    float zmin = __uint_as_float(minmax[0]);
    float zmax = __uint_as_float(minmax[1]);
    bool invalid = (z <= zmin) || (z >= zmax);
    normal1[pix * 3 + 0] = invalid ? 0.f : nx;
    normal1[pix * 3 + 1] = invalid ? 0.f : ny;
    normal1[pix * 3 + 2] = invalid ? 0.f : nz;
}

// ----------------------------------------------- residual/J + WMMA Gram ----
// Per wave: stage 32 augmented-Jacobian rows [J0..J5, r, 0..0] in LDS (16 cols),
// then 8x v_wmma_f32_16x16x4_f32 with A==B register contents accumulates the
// 7x7 Gram block into the top-left of a 16x16 f32 accumulator.
__global__ void icp_accum_kernel(const float* __restrict__ depth0,
                                 const float* __restrict__ depth1,
                                 const float* __restrict__ Kmat,
                                 const float* __restrict__ pose,
                                 const float* __restrict__ normal1,
                                 float* __restrict__ part) {
    __shared__ float tile[WAVES_PER_BLOCK][32][16];
    const int tid  = threadIdx.x;
    const int lane = tid & 31;
    const int wv   = tid >> 5;

    // zero the padding columns once (rows are wave-private, written per lane)
#pragma unroll
    for (int c = 7; c < 16; ++c) tile[wv][lane][c] = 0.0f;

    const float fx = Kmat[0], cx = Kmat[2], fy = Kmat[4], cy = Kmat[5];
    const float R00 = pose[0], R01 = pose[1], R02 = pose[2],  t0 = pose[3];
    const float R10 = pose[4], R11 = pose[5], R12 = pose[6],  t1 = pose[7];
    const float R20 = pose[8], R21 = pose[9], R22 = pose[10], t2 = pose[11];

    const int stride = gridDim.x * blockDim.x;
    int pix = blockIdx.x * blockDim.x + tid;
    const int iters = (NPIX + stride - 1) / stride;

    v8f acc = {0.f, 0.f, 0.f, 0.f, 0.f, 0.f, 0.f, 0.f};

    for (int it = 0; it < iters; ++it, pix += stride) {
        float row0 = 0.f, row1 = 0.f, row2 = 0.f, row3 = 0.f, row4 = 0.f, row5 = 0.f, row6 = 0.f;
        if (pix < NPIX) {
            int j = pix / WW, i = pix - j * WW;
            float d0 = depth0[pix];
            if (d0 > 0.0f) {
                float v0x = ((float)i - cx) / fx * d0;
                float v0y = ((float)j - cy) / fy * d0;
                float v0z = d0;
                float x_ = R00 * v0x + R01 * v0y + R02 * v0z + t0;
                float y_ = R10 * v0x + R11 * v0y + R12 * v0z + t1;
                float z_ = R20 * v0x + R21 * v0y + R22 * v0z + t2;
                if (z_ != 0.0f) {
                    float u_ = x_ / z_ * fx + cx;
                    float v_ = y_ / z_ * fy + cy;
                    bool inview = (u_ > 0.f) && (u_ < (float)(WW - 1)) &&
                                  (v_ > 0.f) && (v_ < (float)(HH - 1));
                    if (inview) {
                        float x0f = floorf(u_), y0f = floorf(v_);
                        float wx = u_ - x0f, wy = v_ - y0f;
                        int x0 = (int)x0f, y0 = (int)y0f;
                        int x1 = x0 + 1, y1 = y0 + 1;
                        float w00 = (1.f - wy) * (1.f - wx);
                        float w01 = (1.f - wy) * wx;
                        float w10 = wy * (1.f - wx);
                        float w11 = wy * wx;

                        float rvx = 0.f, rvy = 0.f, rvz = 0.f;
                        float rnx = 0.f, rny = 0.f, rnz = 0.f;
                        const int   ys[4] = {y0, y0, y1, y1};
                        const int   xs[4] = {x0, x1, x0, x1};
                        const float wsb[4] = {w00, w01, w10, w11};
#pragma unroll
                        for (int k = 0; k < 4; ++k) {
                            int idx = ys[k] * WW + xs[k];
                            float dd = depth1[idx];
                            rvx += wsb[k] * (((float)xs[k] - cx) / fx * dd);
                            rvy += wsb[k] * (((float)ys[k] - cy) / fy * dd);
                            rvz += wsb[k] * dd;
                            rnx += wsb[k] * normal1[idx * 3 + 0];
                            rny += wsb[k] * normal1[idx * 3 + 1];
                            rnz += wsb[k] * normal1[idx * 3 + 2];
                        }
                        bool mask1 = rvz > 0.0f;
                        float dfx = x_ - rvx, dfy = y_ - rvy, dfz = z_ - rvz;
                        float d2 = dfx * dfx + dfy * dfy + dfz * dfz;
                        bool occ = d2 > 0.01f; // norm(diff) > 0.1
                        if (mask1 && !occ) {
                            float res = rnx * dfx + rny * dfy + rnz * dfz;
                            // J_rot = cross(v0t, rn1)
                            row0 = y_ * rnz - z_ * rny;
                            row1 = z_ * rnx - x_ * rnz;
                            row2 = x_ * rny - y_ * rnx;
                            row3 = rnx; row4 = rny; row5 = rnz;
                            row6 = res;
                        }
                    }
                }
            }
        }

        __syncthreads(); // previous tile's reads complete before overwrite
        tile[wv][lane][0] = row0;
        tile[wv][lane][1] = row1;
        tile[wv][lane][2] = row2;
        tile[wv][lane][3] = row3;
        tile[wv][lane][4] = row4;
        tile[wv][lane][5] = row5;
        tile[wv][lane][6] = row6;
        __syncthreads(); // rows visible to all lanes of the wave

        // A(16x4) layout: v0 = K={0|2}, v1 = K={1|3}; lanes 0-15 -> K base 0, 16-31 -> 2.
        // B(4x16) requires the same per-lane data for X^T X, so feed `ab` as both.
        const int half = lane >> 4;
        const int cidx = lane & 15;
#pragma unroll
        for (int q = 0; q < 8; ++q) {
            int r0 = q * 4 + half * 2;
            v2f ab;
            ab.x = tile[wv][r0][cidx];
            ab.y = tile[wv][r0 + 1][cidx];
            acc = __builtin_amdgcn_wmma_f32_16x16x4_f32(
                false, ab, false, ab, (short)0, acc, false, false);
        }
    }

    // D layout: VGPR m, lane n (<16) holds G[m][n]; dump 7x7 partial per wave.
    int slot = blockIdx.x * WAVES_PER_BLOCK + wv;
    if (lane < 7) {
#pragma unroll
        for (int m = 0; m < 7; ++m) {
            part[slot * 49 + m * 7 + lane] = acc[m];
        }
    }
}

// -------------------------------------------- deterministic Gram reduce ----
__global__ void icp_reduce_kernel(const float* __restrict__ part, float* __restrict__ gram) {
    __shared__ float partial[49][8];
    int t = threadIdx.x;
    if (t < 49 * 8) {
        int e = t % 49, chunk = t / 49;
        const int per = NSLOTS / 8;
        int beg = chunk * per, end = beg + per;
        float s = 0.f;
        for (int k = beg; k < end; ++k) s += part[k * 49 + e];
        partial[e][chunk] = s;
    }
    __syncthreads();
    if (t < 49) {
        float s = 0.f;
#pragma unroll
        for (int c = 0; c < 8; ++c) s += partial[t][c];
        gram[t] = s;
    }
}

// --------------------------------------------------- 6x6 solve + exp_se3 ----
__global__ void icp_solve_kernel(float* ws) {
    if (threadIdx.x != 0 || blockIdx.x != 0) return;
    float* gram = ws + GRAM_OFF;

    float M[6][7];
    float trace = 0.f;
    for (int i = 0; i < 6; ++i) trace += gram[i * 7 + i];
    float damp = trace * 0.001f;
    for (int r = 0; r < 6; ++r) {
        for (int c = 0; c < 6; ++c) M[r][c] = gram[r * 7 + c] + ((r == c) ? damp : 0.f);
        M[r][6] = gram[r * 7 + 6];
    }
    // Gauss-Jordan with partial pivoting
    for (int col = 0; col < 6; ++col) {
        int piv = col; float best = fabsf(M[col][col]);
        for (int r = col + 1; r < 6; ++r) {
            float a = fabsf(M[r][col]);
            if (a > best) { best = a; piv = r; }
        }
        if (piv != col) {
            for (int c = 0; c < 7; ++c) { float tmp = M[col][c]; M[col][c] = M[piv][c]; M[piv][c] = tmp; }
        }
        float inv = 1.0f / M[col][col];
        for (int c = col; c < 7; ++c) M[col][c] *= inv;
        for (int r = 0; r < 6; ++r) {
            if (r == col) continue;
            float f = M[r][col];
            for (int c = col; c < 7; ++c) M[r][c] -= f * M[col][c];
        }
    }
    float xi[6];
    for (int k = 0; k < 6; ++k) xi[k] = -M[k][6];

    // exp_se3
    float w0 = xi[0], w1 = xi[1], w2 = xi[2];
    float vv0 = xi[3], vv1 = xi[4], vv2 = xi[5];
    float th2 = w0 * w0 + w1 * w1 + w2 * w2;
    float theta = sqrtf(th2);
    float Rm[3][3] = {{1, 0, 0}, {0, 1, 0}, {0, 0, 1}};
    float Jl[3][3] = {{1, 0, 0}, {0, 1, 0}, {0, 0, 1}};
    if (theta > 1e-8f) {
        float S[3][3] = {{0, -w2, w1}, {w2, 0, -w0}, {-w1, w0, 0}};
        float S2[3][3];
        for (int r = 0; r < 3; ++r)
            for (int c = 0; c < 3; ++c) {
                float s = 0.f;
                for (int k = 0; k < 3; ++k) s += S[r][k] * S[k][c];
                S2[r][c] = s;
            }
        float sn = sinf(theta), cs = cosf(theta);
        float a = sn / theta;
        float b = (1.f - cs) / th2;
        float g = (theta - sn) / (th2 * theta);
        for (int r = 0; r < 3; ++r)
            for (int c = 0; c < 3; ++c) {
                float id = (r == c) ? 1.f : 0.f;
                Rm[r][c] = id + a * S[r][c] + b * S2[r][c];
                Jl[r][c] = id + b * S[r][c] + g * S2[r][c];
            }
    }
    float T[4][4];
    for (int r = 0; r < 3; ++r)
        for (int c = 0; c < 3; ++c) T[r][c] = Rm[r][c];
    T[0][3] = Jl[0][0] * vv0 + Jl[0][1] * vv1 + Jl[0][2] * vv2;
    T[1][3] = Jl[1][0] * vv0 + Jl[1][1] * vv1 + Jl[1][2] * vv2;
    T[2][3] = Jl[2][0] * vv0 + Jl[2][1] * vv1 + Jl[2][2] * vv2;
    T[3][0] = 0.f; T[3][1] = 0.f; T[3][2] = 0.f; T[3][3] = 1.f;

    float P[16];
    for (int k = 0; k < 16; ++k) P[k] = ws[POSE_OFF + k];
    float O[16];
    for (int r = 0; r < 4; ++r)
        for (int c = 0; c < 4; ++c) {
            float s = 0.f;
            for (int k = 0; k < 4; ++k) s += T[r][k] * P[k * 4 + c];
            O[r * 4 + c] = s;
        }
    for (int k = 0; k < 16; ++k) ws[POSE_OFF + k] = O[k];
}

// ------------------------------------------------------------- copy out ----
__global__ void icp_copyout_kernel(const float* __restrict__ ws, float* __restrict__ out) {
    int i = threadIdx.x;
    if (i < 16) out[i] = ws[POSE_OFF + i];
}

// --------------------------------------------------------------- launch ----
extern "C" void kernel_launch(void* const* d_in, const int* in_sizes, int n_in,
                              void* d_out, int out_size, void* d_ws, size_t ws_size,
                              hipStream_t stream) {
    const float* pose10 = (const float*)d_in[0];
    const float* depth0 = (const float*)d_in[1];
    const float* depth1 = (const float*)d_in[2];
    const float* Kmat   = (const float*)d_in[3];

    float*    ws      = (float*)d_ws;
    float*    gram    = ws + GRAM_OFF;
    unsigned* minmax  = ((unsigned*)d_ws) + MINMAX_OFF;
    float*    normal1 = ws + NORM_OFF;
    float*    part    = ws + PART_OFF;
    float*    out     = (float*)d_out;

    icp_init_kernel<<<1, 64, 0, stream>>>(pose10, ws);
    icp_minmax_kernel<<<512, 256, 0, stream>>>(depth1, minmax);
    icp_normal_kernel<<<(NPIX + 255) / 256, 256, 0, stream>>>(depth1, Kmat, minmax, normal1);

    for (int it = 0; it < 3; ++it) {
        icp_accum_kernel<<<ACC_BLOCKS, ACC_THREADS, 0, stream>>>(
            depth0, depth1, Kmat, ws /*pose*/, normal1, part);
        icp_reduce_kernel<<<1, 512, 0, stream>>>(part, gram);
        icp_solve_kernel<<<1, 1, 0, stream>>>(ws);
    }
    icp_copyout_kernel<<<1, 16, 0, stream>>>(ws, out);
}